// SEIRNet_3925600109177
// MI455X (gfx1250) — compile-verified
//
#include <hip/hip_runtime.h>
#include <math.h>

// ---------------------------------------------------------------------------
// SEIRNet: hiddens[t,b,:] is independent of b (identical initial state + scalar
// b_seq[t]) => collapse to a T-step scalar scan + 168 MB streaming broadcast.
// Store-bandwidth bound: 167.8 MB / 23.3 TB/s ~= 7.2 us floor on MI455X.
// ---------------------------------------------------------------------------

typedef __attribute__((ext_vector_type(4))) float    v4f;
typedef __attribute__((ext_vector_type(4))) unsigned v4u;
typedef __attribute__((ext_vector_type(8))) unsigned v8u;

#define SEIR_K  0.2f
#define SEIR_S  0.2f
#define SEIR_I0 5.6e-06f

#define SEIR_B  8192
#define SEIR_F  6
#define SEIR_T  1024

__device__ __forceinline__ float clamp01(float v) {
    return __builtin_amdgcn_fmed3f(v, 0.0f, 1.0f);   // single v_med3_f32
}

// ---------------------------------------------------------------------------
// Phase 1: compute b_seq[t] in parallel (1024 lanes), then one lane runs the
// nonlinear 4-state scan. Writes a 32B record per t into ws:
//   ws[t*8 + 0..3] = h[t][0..3],  ws[t*8 + 4] = out[t]
// ---------------------------------------------------------------------------
__global__ __launch_bounds__(SEIR_T) void seir_scan_kernel(
    const float* __restrict__ X, const float* __restrict__ p_exp,
    const float* __restrict__ q, float* __restrict__ ws)
{
    __shared__ float bseq[SEIR_T];
    const int t = threadIdx.x;

    // X[t, 0, 0:5]; row stride B*F floats, 16B-aligned (B*F*4 % 16 == 0)
    const size_t row = (size_t)t * (SEIR_B * SEIR_F);
    v4f x4 = *(const v4f*)(X + row);
    float x5 = X[row + 4];
    float ss = x4.x * x4.x + x4.y * x4.y + x4.z * x4.z + x4.w * x4.w + x5 * x5;
    bseq[t] = q[0] * powf(sqrtf(ss), p_exp[0]);
    __syncthreads();

    if (t == 0) {
        float p0 = SEIR_I0;
        float p1 = 0.0f;
        float p2 = 1.0f - 2.0f * SEIR_I0;
        float p3 = SEIR_I0;
        float b  = bseq[0];                 // prefetched b for iteration i
        for (int i = 0; i < SEIR_T; ++i) {
            float bn = (i + 1 < SEIR_T) ? bseq[i + 1] : 0.0f;  // hide ds latency
            float d1 = SEIR_K * p0;
            float d2 = p0 * b * p2;
            float d3 = SEIR_S * p3;
            float h0 = clamp01(p0 + d3 - d1);
            float h1 = clamp01(p1 + d1);
            float h2 = clamp01(p2 - d2);
            float h3 = clamp01(p3 + d2 - d3);
            v4f h = {h0, h1, h2, h3};
            *(v4f*)(ws + i * 8) = h;
            ws[i * 8 + 4] = h0 + h1;
            p0 = h0; p1 = h1; p2 = h2; p3 = h3;
            b  = bn;
        }
    }
}

// ---------------------------------------------------------------------------
// Phase 2: broadcast. Block (t, chunk) covers 2048 batch elements of step t.
// Wave 0 TDM-loads the 8-float record ws[t*8..] into LDS (tensor_load_to_lds,
// D# per CDNA5 ISA), then all 256 threads stream
// non-temporal b128 stores (40 KB/block, 4096 blocks).
// ---------------------------------------------------------------------------
__global__ __launch_bounds__(256) void seir_broadcast_kernel(
    const float* __restrict__ ws, float* __restrict__ out)
{
    __shared__ float tile[8];
    const int t     = blockIdx.x;
    const int bbase = blockIdx.y * 2048;

    if (threadIdx.x < 32) {  // wave 0 issues the TDM descriptor
        // Flat shared pointers carry the LDS byte offset in their low 32 bits
        // (aperture mapping, ISA 10.2), which is what D#.lds_addr wants.
        unsigned lds_addr = (unsigned)(size_t)(void*)tile;
        unsigned long long ga = (unsigned long long)(ws + (size_t)t * 8);

        // D# group 0: count=1, lds_addr, global_addr[56:0], type=2
        v4u g0;
        g0[0] = 1u;                                     // count=1, user mode
        g0[1] = lds_addr;                               // LDS dest (bytes)
        g0[2] = (unsigned)ga;                           // global_addr[31:0]
        g0[3] = ((unsigned)(ga >> 32) & 0x01FFFFFFu)    // global_addr[56:32]
                | 0x80000000u;                          // type=2 ("image")
        // D# group 1: data_size=4B, tensor 8x1, tile 8x1, dim0 stride 8
        v8u g1;
        g1[0] = 2u << 16;   // workgroup_mask=0 (not in cluster), data_size=2->4B
        g1[1] = 8u << 16;   // tensor_dim0[15:0]=8 (bits 63:48)
        g1[2] = 1u << 16;   // tensor_dim1[15:0]=1 (bits 111:80 low half)
        g1[3] = 8u << 16;   // tile_dim0=8 (bits 127:112)
        g1[4] = 1u;         // tile_dim1=1, tile_dim2=0 (unused)
        g1[5] = 8u;         // tensor_dim0_stride[31:0]=8
        g1[6] = 0u;         // stride hi / tensor_dim1_stride lo
        g1[7] = 0u;
        // D# groups 2/3: degenerate upper dims
        v4u g2; g2[0] = 1u; g2[1] = 1u; g2[2] = 8u; g2[3] = 0u;
        v4u g3; g3[0] = 8u; g3[1] = 1u << 16; g3[2] = 0u; g3[3] = 0u;

        asm volatile("tensor_load_to_lds %0, %1, %2, %3"
                     :: "s"(g0), "s"(g1), "s"(g2), "s"(g3)
                     : "memory");
        __builtin_amdgcn_s_wait_tensorcnt(0);
    }
    __syncthreads();

    v4f h = *(const v4f*)tile;   // {h0,h1,h2,h3} for this t
    float o = tile[4];           // h0 + h1

    // hiddens[t, b, 0:4] -> out[((t*B + b)*4 ...)]: one b128 NT store per b.
    float* hp = out + ((size_t)t * SEIR_B + bbase) * 4 + (size_t)threadIdx.x * 4;
    #pragma unroll
    for (int k = 0; k < 8; ++k) {
        __builtin_nontemporal_store(h, (v4f*)(hp + (size_t)k * 256 * 4));
    }

    // outputs[t, b] -> out[T*B*4 + t*B + b]: replicate scalar, b128 NT stores.
    float* op = out + (size_t)SEIR_T * SEIR_B * 4 + (size_t)t * SEIR_B + bbase;
    v4f o4 = {o, o, o, o};
    __builtin_nontemporal_store(o4, (v4f*)(op + (size_t)threadIdx.x * 4));
    __builtin_nontemporal_store(o4, (v4f*)(op + 1024 + (size_t)threadIdx.x * 4));
}

extern "C" void kernel_launch(void* const* d_in, const int* in_sizes, int n_in,
                              void* d_out, int out_size, void* d_ws, size_t ws_size,
                              hipStream_t stream) {
    const float* X     = (const float*)d_in[0];
    const float* p_exp = (const float*)d_in[1];
    const float* q     = (const float*)d_in[2];
    float* outp        = (float*)d_out;
    float* ws          = (float*)d_ws;   // needs T*8*4 = 32 KB

    (void)in_sizes; (void)n_in; (void)out_size; (void)ws_size;

    seir_scan_kernel<<<1, SEIR_T, 0, stream>>>(X, p_exp, q, ws);
    seir_broadcast_kernel<<<dim3(SEIR_T, SEIR_B / 2048), 256, 0, stream>>>(ws, outp);
}